// MultiSimilarityLoss_403726926380
// MI455X (gfx1250) — compile-verified
//
#include <hip/hip_runtime.h>
#include <math.h>
#include <stdint.h>

typedef __attribute__((ext_vector_type(2))) float v2f;
typedef __attribute__((ext_vector_type(8))) float v8f;

#define MSL_D          128
#define MSL_ROWS       32          // rows per workgroup (2 waves x 16)
#define MSL_THREADS    64
#define MSL_BSTRIDE    132         // 528B row pitch: 16B-aligned, bank-spread
#define MSL_STAGE_COLS 32          // columns staged per barrier (2 subtiles)

// ---- CDNA5 async global->LDS copy (ASYNCcnt-tracked), per CDNA5 ISA ----
__device__ __forceinline__ void msl_async_g2l_b128(const float* g, float* l)
{
    // VDST vgpr = LDS byte offset (low 32 bits of generic LDS pointer per ISA
    // aperture truncation); VADDR = 64-bit global address; no SADDR.
    uint32_t lds_off = (uint32_t)(uintptr_t)l;
    uint64_t gaddr   = (uint64_t)(uintptr_t)g;
    asm volatile("global_load_async_to_lds_b128 %0, %1, off"
                 :: "v"(lds_off), "v"(gaddr)
                 : "memory");
}
__device__ __forceinline__ void msl_wait_async0()
{
    asm volatile("s_wait_asynccnt 0x0" ::: "memory");
}

// One 16x16 f32 sub-tile of X*X^T, A held in registers, B fragments from LDS.
__device__ __forceinline__ v8f msl_tile_gemm(const v2f* __restrict__ av,
                                             const float* __restrict__ Brow)
{
    v8f acc = {0.f, 0.f, 0.f, 0.f, 0.f, 0.f, 0.f, 0.f};
#pragma unroll
    for (int k4 = 0; k4 < MSL_D / 4; ++k4) {
        v2f b = *(const v2f*)(Brow + k4 * 4);
        acc = __builtin_amdgcn_wmma_f32_16x16x4_f32(false, av[k4], false, b,
                                                    (short)0, acc, false, false);
    }
    return acc;
}

__global__ __launch_bounds__(MSL_THREADS)
void msl_kernel(const float* __restrict__ X, const int* __restrict__ T,
                float* __restrict__ out, int n)
{
    __shared__ float ldsB[2][MSL_STAGE_COLS * MSL_BSTRIDE];

    const int tid  = threadIdx.x;
    const int lane = tid & 31;
    const int wave = tid >> 5;          // 0..1
    const int half = lane >> 4;         // M +0/+8 half and K-pair select
    const int ln   = lane & 15;
    const int myRow0 = blockIdx.x * MSL_ROWS + wave * 16;
    const int numStages = n / MSL_STAGE_COLS;

    // A fragments for this wave's 16-row strip: resident in VGPRs for both passes.
    v2f av[MSL_D / 4];
    {
        const float* Ag = X + (size_t)(myRow0 + ln) * MSL_D + 2 * half;
#pragma unroll
        for (int k4 = 0; k4 < MSL_D / 4; ++k4) av[k4] = *(const v2f*)(Ag + k4 * 4);
    }

    int trow[8];
#pragma unroll
    for (int r = 0; r < 8; ++r) trow[r] = T[myRow0 + r + 8 * half];

    // ================= pass 1: row-wise min over positive pairs =================
    float rmin[8];
#pragma unroll
    for (int r = 0; r < 8; ++r) rmin[r] = __builtin_inff();

    // prologue: stage 0
    for (int j = tid; j < MSL_STAGE_COLS * (MSL_D / 4); j += MSL_THREADS) {
        int c = j >> 5, kq = j & 31;   // D/4 == 32
        msl_async_g2l_b128(X + (size_t)c * MSL_D + kq * 4,
                           &ldsB[0][c * MSL_BSTRIDE + kq * 4]);
    }
    msl_wait_async0();
    __syncthreads();

    for (int st = 0; st < numStages; ++st) {
        const float* cur = ldsB[st & 1];
        if (st + 1 < numStages) {
            const int nb = (st + 1) & 1;
            const size_t colBase = (size_t)(st + 1) * MSL_STAGE_COLS;
            for (int j = tid; j < MSL_STAGE_COLS * (MSL_D / 4); j += MSL_THREADS) {
                int c = j >> 5, kq = j & 31;
                msl_async_g2l_b128(X + (colBase + c) * MSL_D + kq * 4,
                                   &ldsB[nb][c * MSL_BSTRIDE + kq * 4]);
            }
        }
#pragma unroll
        for (int s = 0; s < 2; ++s) {
            const float* Brow = cur + (s * 16 + ln) * MSL_BSTRIDE + 2 * half;
            v8f acc = msl_tile_gemm(av, Brow);
            int tcol = T[st * MSL_STAGE_COLS + s * 16 + ln];
#pragma unroll
            for (int r = 0; r < 8; ++r) {
                float v = acc[r];
                if (trow[r] == tcol && v < 0.9f) rmin[r] = fminf(rmin[r], v);
            }
        }
        msl_wait_async0();
        __syncthreads();
    }
#pragma unroll
    for (int m = 1; m < 16; m <<= 1) {
#pragma unroll
        for (int r = 0; r < 8; ++r)
            rmin[r] = fminf(rmin[r], __shfl_xor(rmin[r], m, 32));
    }

    // ================= pass 2: masked exponential sums =================
    float psum[8], nsum[8], nflag[8];
#pragma unroll
    for (int r = 0; r < 8; ++r) { psum[r] = 0.f; nsum[r] = 0.f; nflag[r] = 0.f; }
    float lps = 0.f, lpc = 0.f, lns = 0.f, lnc = 0.f;   // last-row stats
    const int lastRow = n - 1;

    for (int j = tid; j < MSL_STAGE_COLS * (MSL_D / 4); j += MSL_THREADS) {
        int c = j >> 5, kq = j & 31;
        msl_async_g2l_b128(X + (size_t)c * MSL_D + kq * 4,
                           &ldsB[0][c * MSL_BSTRIDE + kq * 4]);
    }
    msl_wait_async0();
    __syncthreads();

    for (int st = 0; st < numStages; ++st) {
        const float* cur = ldsB[st & 1];
        if (st + 1 < numStages) {
            const int nb = (st + 1) & 1;
            const size_t colBase = (size_t)(st + 1) * MSL_STAGE_COLS;
            for (int j = tid; j < MSL_STAGE_COLS * (MSL_D / 4); j += MSL_THREADS) {
                int c = j >> 5, kq = j & 31;
                msl_async_g2l_b128(X + (colBase + c) * MSL_D + kq * 4,
                                   &ldsB[nb][c * MSL_BSTRIDE + kq * 4]);
            }
        }
#pragma unroll
        for (int s = 0; s < 2; ++s) {
            const float* Brow = cur + (s * 16 + ln) * MSL_BSTRIDE + 2 * half;
            v8f acc = msl_tile_gemm(av, Brow);
            int tcol = T[st * MSL_STAGE_COLS + s * 16 + ln];
#pragma unroll
            for (int r = 0; r < 8; ++r) {
                float v    = acc[r];
                bool eq    = (trow[r] == tcol);
                bool pos   = eq && (v < 0.9f);      // diagonal self-excluded (v~1)
                bool neg   = (!eq) && ((v + 0.5f) > rmin[r]) && (v > 0.1f);
                bool last  = (myRow0 + r + 8 * half) == lastRow;
                if (pos) {
                    float d = v - 0.9f;
                    psum[r] += __expf(-2.0f * (v - 0.5f) + d * d);  // ep = 1.0
                    if (last) { lps += v; lpc += 1.f; }
                }
                if (neg) {
                    float d = 0.1f - v;
                    nsum[r] += __expf(50.0f * (v - 0.5f) + d * d);
                    nflag[r] = 1.f;
                    if (last) { lns += v; lnc += 1.f; }
                }
            }
        }
        msl_wait_async0();
        __syncthreads();
    }

#pragma unroll
    for (int m = 1; m < 16; m <<= 1) {
#pragma unroll
        for (int r = 0; r < 8; ++r) {
            psum[r]  += __shfl_xor(psum[r], m, 32);
            nsum[r]  += __shfl_xor(nsum[r], m, 32);
            nflag[r]  = fmaxf(nflag[r], __shfl_xor(nflag[r], m, 32));
        }
    }

    // per-row loss; identical across the 16 lanes of a half -> combine halves
    float localLoss = 0.f, localSkip = 0.f;
#pragma unroll
    for (int r = 0; r < 8; ++r) {
        if (nflag[r] > 0.f)
            localLoss += 0.5f * log1pf(psum[r]) + 0.02f * log1pf(nsum[r]);
        else
            localSkip += 1.f;
    }
    localLoss += __shfl_xor(localLoss, 16, 32);
    localSkip += __shfl_xor(localSkip, 16, 32);
    if (lane == 0) {
        atomicAdd(&out[0], localLoss / (float)n);   // loss
        atomicAdd(&out[1], localSkip / (float)n);   // prec
    }

    // last-row pair means (owned by exactly one wave in the grid)
#pragma unroll
    for (int m = 1; m < 32; m <<= 1) {
        lps += __shfl_xor(lps, m, 32);
        lpc += __shfl_xor(lpc, m, 32);
        lns += __shfl_xor(lns, m, 32);
        lnc += __shfl_xor(lnc, m, 32);
    }
    if (myRow0 == n - 16 && lane == 0) {
        out[2] = lps / fmaxf(lpc, 1.f);
        out[3] = lns / fmaxf(lnc, 1.f);
    }
}

extern "C" void kernel_launch(void* const* d_in, const int* in_sizes, int n_in,
                              void* d_out, int out_size, void* d_ws, size_t ws_size,
                              hipStream_t stream) {
    const float* X = (const float*)d_in[0];   // [n,128] f32, L2-normalized
    const int*   T = (const int*)d_in[1];     // [n] int32
    float* out = (float*)d_out;               // 4 f32: loss, prec, mean_pos, mean_neg
    const int n = in_sizes[1];

    hipMemsetAsync(d_out, 0, (size_t)out_size * sizeof(float), stream);

    const int blocks = n / MSL_ROWS;          // n=8192 -> 256 workgroups
    msl_kernel<<<blocks, MSL_THREADS, 0, stream>>>(X, T, out, n);
}